// SPDSRU_39402029974125
// MI455X (gfx1250) — compile-verified
//
#include <hip/hip_runtime.h>
#include <hip/hip_bf16.h>

typedef __attribute__((ext_vector_type(2))) float v2f;
typedef __attribute__((ext_vector_type(8))) float v8f;

// ---------------------------------------------------------------------------
// Implicit-GEMM 7x7 conv via V_WMMA_F32_16X16X4_F32.
// One wave computes a 16(pixel) x 16(out-channel, padded) tile.
// K is permuted as k = (ci,r)*8 + c (7 real columns + 1 zero pad) so the
// inner-loop K decode is just shifts/masks. The wave's 22x7xCIN receptive
// field is staged in LDS once; the K-loop reads A and B from LDS only.
// CIN==3 input is NHWC (reference x layout); CIN==5 input is NCHW.
// ---------------------------------------------------------------------------
template <int CIN, int COUT, int H, int W>
__global__ __launch_bounds__(256) void conv7x7_wmma(
    const float* __restrict__ in, const float* __restrict__ wgt,
    const float* __restrict__ bias, float* __restrict__ out, int nframes)
{
    constexpr int NPAIR = CIN * 7;        // (ci, r) pairs
    constexpr int KP    = NPAIR * 8;      // padded K (multiple of 8)
    constexpr int TS    = NPAIR * 22;     // staged tile elements per wave
    constexpr int TPR   = W / 16;         // 16-pixel tiles per row
    __shared__ float Wlds[KP][16];
    __shared__ float ATile[8][TS];

    // Weight tile in permuted-K order (block-wide fill).
    for (int idx = threadIdx.x; idx < KP * 16; idx += blockDim.x) {
        int k = idx >> 4, co = idx & 15;
        int pair = k >> 3, cc = k & 7;
        int ci = pair / 7, r = pair - ci * 7;
        float v = 0.f;
        if (cc < 7 && co < COUT)
            v = wgt[((co * CIN + ci) * 7 + r) * 7 + cc];
        Wlds[k][co] = v;
    }

    const int wave = threadIdx.x >> 5;
    const int lane = threadIdx.x & 31;
    const int half = lane >> 4;           // K-pair selector within fragment
    const int lm   = lane & 15;           // M for A-frag, N for B/C-frag

    long tile   = (long)blockIdx.x * 8 + wave;
    long ntiles = (long)nframes * H * TPR;
    bool valid  = tile < ntiles;
    if (!valid) tile = 0;                 // clamp; stores are guarded below
    int n  = (int)(tile / (H * TPR));
    int rp = (int)(tile % (H * TPR));
    int py = rp / TPR;
    int x0 = (rp % TPR) * 16;

    // Stage receptive field: columns x0-3 .. x0+18, rows py-3 .. py+3, CIN ch.
    for (int idx = lane; idx < TS; idx += 32) {
        int xx = idx % 22; int t2 = idx / 22;   // t2 = pair = ci*7 + r
        int r = t2 % 7, ci = t2 / 7;
        int iy = py + r - 3, ix = x0 + xx - 3;
        float v = 0.f;
        if (iy >= 0 && iy < H && ix >= 0 && ix < W) {
            if (CIN == 3)
                v = in[((long)(n * H + iy) * W + ix) * 3 + ci];
            else
                v = in[(((long)n * CIN + ci) * H + iy) * W + ix];
        }
        ATile[wave][idx] = v;
    }
    __syncthreads();

    const float* AT = &ATile[wave][0];
    v8f acc = {};
#pragma unroll 4
    for (int k0 = 0; k0 < KP; k0 += 4) {
        int ka   = k0 + 2 * half;
        int pair = ka >> 3;
        int cc   = ka & 7;                // in {0,2,4,6}
        int ab   = pair * 22 + lm + cc;
        v2f a, b;
        a.x = AT[ab];                                 // cc   < 7 always
        a.y = (cc != 6) ? AT[ab + 1] : 0.f;           // pad column
        b.x = Wlds[ka][lm];
        b.y = Wlds[ka + 1][lm];
        acc = __builtin_amdgcn_wmma_f32_16x16x4_f32(
            false, a, false, b, (short)0, acc, false, false);
    }

    if (valid && lm < COUT) {             // this lane holds out-channel N=lm
        float bs = bias[lm];
#pragma unroll
        for (int v = 0; v < 8; ++v) {
            int m  = v + 8 * half;        // pixel index within tile
            int px = x0 + m;
            out[(((long)n * COUT + lm) * H + py) * W + px] = acc[v] + bs;
        }
    }
}

// ---------------------------------------------------------------------------
// Deterministic BN statistics: fixed-slice partial sums + fixed-order combine.
// ---------------------------------------------------------------------------
__global__ __launch_bounds__(256) void bn_reduce_partial(
    const float* __restrict__ y, int C, int HW, long L, int P,
    float* __restrict__ partials)
{
    int c = blockIdx.x / P, p = blockIdx.x % P;
    long chunk = (L + P - 1) / P;
    long j0 = (long)p * chunk, j1 = j0 + chunk; if (j1 > L) j1 = L;
    float s = 0.f, ss = 0.f;
    for (long j = j0 + threadIdx.x; j < j1; j += blockDim.x) {
        long nf = j / HW; long pos = j - nf * HW;
        float v = y[(nf * C + c) * (long)HW + pos];
        s += v; ss += v * v;
    }
    __shared__ float red0[256], red1[256];
    red0[threadIdx.x] = s; red1[threadIdx.x] = ss;
    __syncthreads();
    for (int w = 128; w > 0; w >>= 1) {
        if ((int)threadIdx.x < w) {
            red0[threadIdx.x] += red0[threadIdx.x + w];
            red1[threadIdx.x] += red1[threadIdx.x + w];
        }
        __syncthreads();
    }
    if (threadIdx.x == 0) {
        partials[(c * P + p) * 2 + 0] = red0[0];
        partials[(c * P + p) * 2 + 1] = red1[0];
    }
}

__global__ void bn_finalize(const float* __restrict__ partials, int C, int P,
                            float invL, float* __restrict__ mv)
{
    int c = threadIdx.x;
    if (c < C) {
        float s = 0.f, ss = 0.f;
        for (int p = 0; p < P; ++p) {
            s  += partials[(c * P + p) * 2 + 0];
            ss += partials[(c * P + p) * 2 + 1];
        }
        float mu  = s * invL;
        float var = ss * invL - mu * mu;
        mv[2 * c]     = mu;
        mv[2 * c + 1] = rsqrtf(var + 1e-5f);
    }
}

// ---------------------------------------------------------------------------
// BN + ReLU + 2x2 maxpool. Frame selection n = b*fs + fo lets pool2 touch
// only the 8 last-time-step frames that feed the recurrence.
// ---------------------------------------------------------------------------
__global__ __launch_bounds__(256) void bn_relu_pool(
    const float* __restrict__ y, const float* __restrict__ mv,
    const float* __restrict__ g, const float* __restrict__ bb,
    float* __restrict__ out, int NSEL, int fs, int fo, int C, int H, int W)
{
    int OH = H / 2, OW = W / 2;
    long total = (long)NSEL * C * OH * OW;
    for (long i = (long)blockIdx.x * blockDim.x + threadIdx.x; i < total;
         i += (long)gridDim.x * blockDim.x) {
        int ox = (int)(i % OW); long t = i / OW;
        int oy = (int)(t % OH); t /= OH;
        int c  = (int)(t % C);  int b = (int)(t / C);
        int n  = b * fs + fo;
        float mu = mv[2 * c], inv = mv[2 * c + 1], gg = g[c], bv = bb[c];
        float best = -1e30f;
#pragma unroll
        for (int dy = 0; dy < 2; ++dy)
#pragma unroll
            for (int dx = 0; dx < 2; ++dx) {
                float v = y[(((long)n * C + c) * H + (2 * oy + dy)) * W + (2 * ox + dx)];
                v = (v - mu) * inv * gg + bv;
                v = v > 0.f ? v : 0.f;
                best = v > best ? v : best;
            }
        out[(((long)b * C + c) * OH + oy) * OW + ox] = best;
    }
}

// ---------------------------------------------------------------------------
// Covariance feature via WMMA: G = F1pad * F1pad^T (16x16 padded, K=1200).
// Under the CDNA5 f32 fragment layout the A and B fragments of a symmetric
// product are identical registers, so the same v2f is passed twice.
// cov = G - R*m m^T + beta^2 m m^T, then 8x8 homogeneous augmentation.
// ---------------------------------------------------------------------------
__global__ __launch_bounds__(32) void cov_wmma(const float* __restrict__ h2,
                                               float* __restrict__ feat)
{
    __shared__ float F[7 * 30 * 40];
    __shared__ float G[16][16];
    __shared__ float S[7];
    __shared__ float red[32];
    const float* src = h2 + (long)blockIdx.x * 8400;
    for (int i = threadIdx.x; i < 8400; i += 32) F[i] = src[i];
    __syncthreads();

    const int lane = threadIdx.x, half = lane >> 4, lm = lane & 15;
    v8f acc = {};
    for (int k0 = 0; k0 < 1200; k0 += 4) {
        int ka = k0 + 2 * half;
        v2f a;
#pragma unroll
        for (int t = 0; t < 2; ++t) {
            int k = ka + t;
            float v = 0.f;
            if (lm < 7) {            // rows >= 7 are zero padding
                int flat = lm * 1200 + k;
                int c  = flat % 7;
                int hh = (flat % 210) / 7;
                int w  = flat / 210;
                v = F[(c * 30 + hh) * 40 + w];
            }
            if (t == 0) a.x = v; else a.y = v;
        }
        acc = __builtin_amdgcn_wmma_f32_16x16x4_f32(
            false, a, false, a, (short)0, acc, false, false);
    }
#pragma unroll
    for (int v = 0; v < 8; ++v) G[v + 8 * half][lm] = acc[v];

    for (int i = 0; i < 7; ++i) {        // fixed-order row sums
        float p = 0.f;
        for (int j = lane; j < 1200; j += 32) {
            int flat = i * 1200 + j;
            int c = flat % 7, hh = (flat % 210) / 7, w = flat / 210;
            p += F[(c * 30 + hh) * 40 + w];
        }
        red[lane] = p;
        __syncthreads();
        if (lane == 0) { float s = 0.f; for (int l = 0; l < 32; ++l) s += red[l]; S[i] = s; }
        __syncthreads();
    }

    const float R = 1200.f, B = 0.3f;
    for (int e = lane; e < 64; e += 32) {
        int i = e >> 3, j = e & 7;
        float mi = (i < 7) ? S[i] / R : 0.f;
        float mj = (j < 7) ? S[j] / R : 0.f;
        float v;
        if (i < 7 && j < 7)      v = G[i][j] - R * mi * mj + B * B * mi * mj;
        else if (i < 7)          v = B * mi;
        else if (j < 7)          v = B * mj;
        else                     v = 1.f;
        feat[blockIdx.x * 64 + e] = v;
    }
}

// ---------------------------------------------------------------------------
// Recurrence (M stays M0 = 1e-5*I, so nus terms collapse to scalars),
// Cayley transforms via per-thread 8x8 Gauss-Jordan with partial pivoting,
// then tril36 -> batch BN -> classifier. One 64-thread block; tiny.
// ---------------------------------------------------------------------------
__global__ __launch_bounds__(64) void recur_head(
    const float* __restrict__ feat,
    const float* __restrict__ kr,  const float* __restrict__ kt,
    const float* __restrict__ kphi, const float* __restrict__ ks_,
    const float* __restrict__ br,  const float* __restrict__ bt,
    const float* __restrict__ by,
    const float* __restrict__ b3g, const float* __restrict__ b3b,
    const float* __restrict__ wcls, const float* __restrict__ bcls,
    float* __restrict__ out)
{
    __shared__ float Q[3][5][8][8];       // cayley(br), cayley(bt), cayley(by)
    __shared__ float h[8][8][8];
    __shared__ float Rt[8][8];
    __shared__ float T1[8][8][8], T2[8][8][8];
    __shared__ float z[8][36];
    __shared__ float mz[36], vz[36];
    const int tid = threadIdx.x;
    const float EPSf = 1e-10f;
    const float AL[5] = {0.01f, 0.25f, 0.5f, 0.9f, 0.99f};

    for (int i = tid; i < 512; i += 64) h[i >> 6][(i >> 3) & 7][i & 7] = feat[i];

    if (tid < 15) {                       // 15 independent Cayley solves
        int which = tid / 5, d = tid % 5;
        const float* bv = (which == 0 ? br : which == 1 ? bt : by) + d * 28;
        float Sm[8][8];
        for (int i = 0; i < 8; ++i) for (int j = 0; j < 8; ++j) Sm[i][j] = 0.f;
        for (int r = 1; r < 8; ++r)
            for (int c = 0; c < r; ++c) Sm[r][c] = bv[r - 1 + c];
        for (int r = 0; r < 8; ++r)
            for (int c = r + 1; c < 8; ++c) Sm[r][c] = -Sm[c][r];
        float A[8][8], Bm[8][8];
        for (int i = 0; i < 8; ++i) for (int j = 0; j < 8; ++j) {
            float I = (i == j) ? 1.f : 0.f;
            A[i][j]  = I - Sm[i][j];
            Bm[i][j] = I + Sm[i][j];
        }
        for (int col = 0; col < 8; ++col) {
            int piv = col; float mx = fabsf(A[col][col]);
            for (int rr = col + 1; rr < 8; ++rr) {
                float v = fabsf(A[rr][col]); if (v > mx) { mx = v; piv = rr; }
            }
            if (piv != col)
                for (int j = 0; j < 8; ++j) {
                    float t = A[col][j];  A[col][j]  = A[piv][j];  A[piv][j]  = t;
                    float u = Bm[col][j]; Bm[col][j] = Bm[piv][j]; Bm[piv][j] = u;
                }
            float inv = 1.f / A[col][col];
            for (int j = 0; j < 8; ++j) { A[col][j] *= inv; Bm[col][j] *= inv; }
            for (int rr = 0; rr < 8; ++rr) if (rr != col) {
                float f = A[rr][col];
                for (int j = 0; j < 8; ++j) {
                    A[rr][j]  -= f * A[col][j];
                    Bm[rr][j] -= f * Bm[col][j];
                }
            }
        }
        for (int i = 0; i < 8; ++i) for (int j = 0; j < 8; ++j)
            Q[which][d][i][j] = Bm[i][j];
    }
    __syncthreads();

    const int i = tid >> 3, j = tid & 7;
    for (int d = 0; d < 5; ++d) {
        float swr = 0.f;
        for (int a = 0; a < 5; ++a) { float w = kr[d * 5 + a]; swr += w * w; }
        float cY = 1e-5f * (swr / (swr + EPSf));            // Yt = cY * I
        float ktv = kt[d], kpv = kphi[d];
        float at = ktv * ktv / (ktv * ktv + kpv * kpv + EPSf);
        float sws = 0.f;
        for (int a = 0; a < 5; ++a) { float w = ks_[d * 5 + a]; sws += w * w; }
        float c1 = 0.f, c2 = 0.f;                           // St = c1*I + c2*Phit
        for (int a = 0; a < 5; ++a) {
            float wn = ks_[d * 5 + a] * ks_[d * 5 + a] / (sws + EPSf);
            c1 += wn * (1.f - AL[a]) * 1e-5f;
            c2 += wn * AL[a];
        }

        { float s = 0.f;                                    // Rt = cY * Qbr Qbr^T
          for (int k = 0; k < 8; ++k) s += Q[0][d][i][k] * Q[0][d][j][k];
          Rt[i][j] = cY * s; }
        __syncthreads();
        for (int b = 0; b < 8; ++b) {                       // T1 = Qbt * X
            float s = 0.f;
            for (int k = 0; k < 8; ++k) {
                float X = (1.f - at) * h[b][k][j] + at * Rt[k][j];
                s += Q[1][d][i][k] * X;
            }
            T1[b][i][j] = s;
        }
        __syncthreads();
        for (int b = 0; b < 8; ++b) {                       // T2 = Phit = T1 * Qbt^T
            float s = 0.f;
            for (int k = 0; k < 8; ++k) s += T1[b][i][k] * Q[1][d][j][k];
            T2[b][i][j] = s;
        }
        __syncthreads();
        for (int b = 0; b < 8; ++b) {                       // T1 = Qby * St
            float s = 0.f;
            for (int k = 0; k < 8; ++k) s += Q[2][d][i][k] * T2[b][k][j];
            T1[b][i][j] = c1 * Q[2][d][i][j] + c2 * s;
        }
        __syncthreads();
        for (int b = 0; b < 8; ++b) {                       // h = T1 * Qby^T
            float s = 0.f;
            for (int k = 0; k < 8; ++k) s += T1[b][i][k] * Q[2][d][j][k];
            h[b][i][j] = s;
        }
        __syncthreads();
    }

    for (int e = tid; e < 8 * 36; e += 64) {                // tril extraction
        int b = e / 36, t = e % 36;
        int r = 0; while ((r + 1) * (r + 2) / 2 <= t) ++r;
        int c = t - r * (r + 1) / 2;
        z[b][t] = h[b][r][c];
    }
    __syncthreads();
    if (tid < 36) {
        float s = 0.f, ss = 0.f;
        for (int b = 0; b < 8; ++b) { float v = z[b][tid]; s += v; ss += v * v; }
        float mu = s * 0.125f, var = ss * 0.125f - mu * mu;
        mz[tid] = mu; vz[tid] = rsqrtf(var + 1e-5f);
    }
    __syncthreads();
    for (int e = tid; e < 8 * 36; e += 64) {
        int b = e / 36, t = e % 36;
        z[b][t] = (z[b][t] - mz[t]) * vz[t] * b3g[t] + b3b[t];
    }
    __syncthreads();
    for (int e = tid; e < 88; e += 64) {
        int b = e / 11, o = e % 11;
        float s = bcls[o];
        for (int t = 0; t < 36; ++t) s += z[b][t] * wcls[o * 36 + t];
        out[b * 11 + o] = s;
    }
}

// ---------------------------------------------------------------------------
extern "C" void kernel_launch(void* const* d_in, const int* in_sizes, int n_in,
                              void* d_out, int out_size, void* d_ws, size_t ws_size,
                              hipStream_t stream)
{
    const float* x    = (const float*)d_in[0];
    const float* c1w  = (const float*)d_in[1];
    const float* c1b  = (const float*)d_in[2];
    const float* b1g  = (const float*)d_in[3];
    const float* b1b  = (const float*)d_in[4];
    const float* c2w  = (const float*)d_in[5];
    const float* c2b  = (const float*)d_in[6];
    const float* b2g  = (const float*)d_in[7];
    const float* b2b  = (const float*)d_in[8];
    const float* kr   = (const float*)d_in[9];
    const float* kt   = (const float*)d_in[10];
    const float* kphi = (const float*)d_in[11];
    const float* ks_  = (const float*)d_in[12];
    const float* br   = (const float*)d_in[13];
    const float* bt   = (const float*)d_in[14];
    const float* by   = (const float*)d_in[15];
    const float* b3g  = (const float*)d_in[16];
    const float* b3b  = (const float*)d_in[17];
    const float* wcls = (const float*)d_in[18];
    const float* bcls = (const float*)d_in[19];
    float* out = (float*)d_out;
    float* ws  = (float*)d_ws;
    (void)in_sizes; (void)n_in; (void)out_size; (void)ws_size;

    // workspace layout (floats)
    float* Y1   = ws;                  // 128*5*120*160 = 12,288,000
    float* H1   = Y1 + 12288000;       // 128*5* 60* 80 =  3,072,000
    float* Y2   = H1 + 3072000;        // 128*7* 60* 80 =  4,300,800 (5,376,000 reserved)
    float* H2   = Y2 + 5376000;        //   8*7* 30* 40 =     67,200
    float* P1   = H2 + 67200;          // 5*64*2 partials
    float* P2   = P1 + 640;            // 7*64*2 partials
    float* MV1  = P2 + 896;            // mean/invstd per ch
    float* MV2  = MV1 + 10;
    float* FEAT = MV2 + 14;            // 8 frames * 8*8

    // conv1: 128*120*(160/16) = 153600 tiles, 8 waves/block
    conv7x7_wmma<3, 5, 120, 160><<<19200, 256, 0, stream>>>(x, c1w, c1b, Y1, 128);
    bn_reduce_partial<<<5 * 64, 256, 0, stream>>>(Y1, 5, 120 * 160,
                                                  (long)128 * 120 * 160, 64, P1);
    bn_finalize<<<1, 64, 0, stream>>>(P1, 5, 64, 1.f / (128.f * 120 * 160), MV1);
    bn_relu_pool<<<12000, 256, 0, stream>>>(Y1, MV1, b1g, b1b, H1,
                                            128, 1, 0, 5, 120, 160);

    // conv2: 128*60*(80/16) = 38400 tiles
    conv7x7_wmma<5, 7, 60, 80><<<4800, 256, 0, stream>>>(H1, c2w, c2b, Y2, 128);
    bn_reduce_partial<<<7 * 64, 256, 0, stream>>>(Y2, 7, 60 * 80,
                                                  (long)128 * 60 * 80, 64, P2);
    bn_finalize<<<1, 64, 0, stream>>>(P2, 7, 64, 1.f / (128.f * 60 * 80), MV2);
    // pool2 only for the 8 frames (n = 16b + 15) that feed the recurrence
    bn_relu_pool<<<300, 256, 0, stream>>>(Y2, MV2, b2g, b2b, H2,
                                          8, 16, 15, 7, 60, 80);

    cov_wmma<<<8, 32, 0, stream>>>(H2, FEAT);
    recur_head<<<1, 64, 0, stream>>>(FEAT, kr, kt, kphi, ks_, br, bt, by,
                                     b3g, b3b, wcls, bcls, out);
}